// SO2Linear_13254269075600
// MI455X (gfx1250) — compile-verified
//
#include <hip/hip_runtime.h>

// ---------------------------------------------------------------------------
// SO2 Linear (L=6, C_IN=C_OUT=64, N=8192) for gfx1250 using V_WMMA_F32_16X16X4_F32
// ---------------------------------------------------------------------------

#define L_MAX   6
#define MS      49          // (L+1)^2 input/output m-rows
#define NE      413         // total (M_out, M_in) edges
#define CIN     64
#define COUT    64
#define NROWS   8192
#define ROWLEN  (MS * CIN)  // 3136 floats per n
#define XS_STRIDE 68        // padded LDS stride (floats) for X tile
#define WS_STRIDE 68        // padded LDS stride (floats) for transposed W tile

typedef __attribute__((ext_vector_type(2))) float v2f;
typedef __attribute__((ext_vector_type(8))) float v8f;

// ---------------------------------------------------------------------------
// Compile-time edge tables (CSR grouped by output row). The reference index
// arrays are pure functions of L=6, so we bake them in as __constant__ data.
// ---------------------------------------------------------------------------
struct Tables {
  int   row_start[MS + 1];
  int   mi[NE];     // input m-row index
  int   wi[NE];     // weight slot index
  float sgn[NE];    // +-1
};

constexpr int imin_(int a, int b) { return a < b ? a : b; }

constexpr Tables build_tables() {
  Tables T{};
  // weight-slot base per (lo, li), matching reference loop order
  int wbase[L_MAX + 1][L_MAX + 1] = {};
  int w = 0;
  for (int lo = 0; lo <= L_MAX; ++lo)
    for (int li = 0; li <= L_MAX; ++li) {
      wbase[lo][li] = w;
      w += 2 * imin_(lo, li) + 1;
    }
  int e = 0, row = 0;
  for (int lo = 0; lo <= L_MAX; ++lo) {
    for (int m = -lo; m <= lo; ++m) {   // output row index = lo*lo + lo + m
      T.row_start[row] = e;
      const int a = m < 0 ? -m : m;
      for (int li = a; li <= L_MAX; ++li) {
        const int mn   = imin_(lo, li);
        const int base = wbase[lo][li];
        const int mic  = li * li + li;  // d2i(li, 0, 0)
        if (m == 0) {
          T.mi[e] = mic;     T.wi[e] = base + mn;     T.sgn[e] =  1.0f; ++e;
        } else if (m > 0) {
          // mw = -m contributes (mo=+m, mi=-m, sign=-1); mw=+m -> (mo=+m, mi=+m, +1)
          T.mi[e] = mic - a; T.wi[e] = base + mn - a; T.sgn[e] = -1.0f; ++e;
          T.mi[e] = mic + a; T.wi[e] = base + mn + a; T.sgn[e] =  1.0f; ++e;
        } else {
          // mw = +a contributes (mo=-a, mi=-a, +1); mw=-a -> (mo=-a, mi=+a, +1)
          T.mi[e] = mic - a; T.wi[e] = base + mn + a; T.sgn[e] =  1.0f; ++e;
          T.mi[e] = mic + a; T.wi[e] = base + mn - a; T.sgn[e] =  1.0f; ++e;
        }
      }
      ++row;
    }
  }
  T.row_start[row] = e;  // == NE
  return T;
}

__constant__ Tables g_tab = build_tables();

// ---------------------------------------------------------------------------
// Kernel: block = 256 threads (8 waves). Block handles 128 n-rows x 1 output
// m-row. Wave w handles n-rows [n0+16w, n0+16w+16) x all 64 output channels
// (4 WMMA C tiles, fp32 accum). Edge loop stages X tile and sign-scaled
// transposed W tile in LDS, then does 16 K-steps of v_wmma_f32_16x16x4_f32.
// ---------------------------------------------------------------------------
__global__ __launch_bounds__(256) void so2_linear_wmma(
    const float* __restrict__ x,       // [N, 49, 64]
    const float* __restrict__ weight,  // [231, 64, 64]
    float* __restrict__ out)           // [N, 49, 64]
{
  __shared__ float Xs[128 * XS_STRIDE];   // X tile, row-major [128][64] padded
  __shared__ float Ws[COUT * WS_STRIDE];  // W tile TRANSPOSED: Ws[d][c] padded

  const int n0   = blockIdx.x * 128;
  const int mo   = blockIdx.y;            // output m-row, uniform per block
  const int tid  = threadIdx.x;
  const int wv   = tid >> 5;              // wave id 0..7 (wave32)
  const int lane = tid & 31;
  const int l    = lane & 15;             // N-col / M-row sub-index
  const int h    = lane >> 4;             // half-wave select

  v8f acc0 = {}, acc1 = {}, acc2 = {}, acc3 = {};

  const int ebeg = g_tab.row_start[mo];
  const int eend = g_tab.row_start[mo + 1];

  for (int e = ebeg; e < eend; ++e) {
    const int   mi = g_tab.mi[e];
    const int   wi = g_tab.wi[e];
    const float sg = g_tab.sgn[e];

    __syncthreads();  // previous iteration's LDS reads complete

    // ---- stage X tile: 128 rows x 64 ch, b128 coalesced loads ----
    const float* xsrc = x + (size_t)n0 * ROWLEN + (size_t)mi * CIN;
#pragma unroll
    for (int j = 0; j < 8; ++j) {
      const int p   = j * 256 + tid;      // 16-byte chunk id
      const int row = p >> 4;
      const int cq  = (p & 15) << 2;
      const float4 v = *(const float4*)(xsrc + (size_t)row * ROWLEN + cq);
      *(float4*)(&Xs[row * XS_STRIDE + cq]) = v;
    }

    // ---- stage W tile transposed with sign folded in ----
    const float* wsrc = weight + (size_t)wi * (CIN * COUT);
#pragma unroll
    for (int j = 0; j < 4; ++j) {
      const int p  = j * 256 + tid;
      const int c  = p >> 4;              // input-channel (K) index
      const int d0 = (p & 15) << 2;       // output-channel base
      const float4 v = *(const float4*)(wsrc + c * COUT + d0);
      Ws[(d0 + 0) * WS_STRIDE + c] = sg * v.x;
      Ws[(d0 + 1) * WS_STRIDE + c] = sg * v.y;
      Ws[(d0 + 2) * WS_STRIDE + c] = sg * v.z;
      Ws[(d0 + 3) * WS_STRIDE + c] = sg * v.w;
    }

    __syncthreads();

    // ---- WMMA K-loop: A 16x4 f32 (lane l holds M=l, K = 2h,2h+1 contiguous)
    //      B 4x16 f32 from transposed W tile (one ds_load_b64, conflict-free)
    const float* Arow = &Xs[(wv * 16 + l) * XS_STRIDE + 2 * h];
    const float* Brow = &Ws[l * WS_STRIDE + 2 * h];
#pragma unroll
    for (int k = 0; k < 16; ++k) {
      const v2f a  = *(const v2f*)(Arow + 4 * k);
      const v2f b0 = *(const v2f*)(Brow + 0 * 16 * WS_STRIDE + 4 * k);
      const v2f b1 = *(const v2f*)(Brow + 1 * 16 * WS_STRIDE + 4 * k);
      const v2f b2 = *(const v2f*)(Brow + 2 * 16 * WS_STRIDE + 4 * k);
      const v2f b3 = *(const v2f*)(Brow + 3 * 16 * WS_STRIDE + 4 * k);
      acc0 = __builtin_amdgcn_wmma_f32_16x16x4_f32(false, a, false, b0,
                                                   (short)0, acc0, false, false);
      acc1 = __builtin_amdgcn_wmma_f32_16x16x4_f32(false, a, false, b1,
                                                   (short)0, acc1, false, false);
      acc2 = __builtin_amdgcn_wmma_f32_16x16x4_f32(false, a, false, b2,
                                                   (short)0, acc2, false, false);
      acc3 = __builtin_amdgcn_wmma_f32_16x16x4_f32(false, a, false, b3,
                                                   (short)0, acc3, false, false);
    }
  }

  // ---- store: C layout VGPR g -> rows (g + 8h), col = lane%16 ----
  float* obase = out + (size_t)(n0 + wv * 16) * ROWLEN + (size_t)mo * COUT + l;
#pragma unroll
  for (int g = 0; g < 8; ++g) {
    float* orow = obase + (size_t)(g + 8 * h) * ROWLEN;
    orow[0]  = acc0[g];
    orow[16] = acc1[g];
    orow[32] = acc2[g];
    orow[48] = acc3[g];
  }
}

// ---------------------------------------------------------------------------
extern "C" void kernel_launch(void* const* d_in, const int* in_sizes, int n_in,
                              void* d_out, int out_size, void* d_ws, size_t ws_size,
                              hipStream_t stream) {
  (void)in_sizes; (void)n_in; (void)out_size; (void)d_ws; (void)ws_size;
  const float* x      = (const float*)d_in[0];  // [8192, 49, 64] f32
  const float* weight = (const float*)d_in[1];  // [1, 231, 64, 64] f32
  // d_in[2..5] (weight_sign, M_in, M_out, weight_index) are compile-time known
  float* out = (float*)d_out;                   // [8192, 49, 64] f32

  dim3 grid(NROWS / 128, MS);
  dim3 block(256);
  so2_linear_wmma<<<grid, block, 0, stream>>>(x, weight, out);
}